// PatchSA_67954972557365
// MI455X (gfx1250) — compile-verified
//
#include <hip/hip_runtime.h>

typedef __attribute__((ext_vector_type(2))) float v2f;
typedef __attribute__((ext_vector_type(4))) float v4f;
typedef __attribute__((ext_vector_type(8))) float v8f;

#define NB      8
#define DIM     256
#define IMG     160
#define HW      (IMG * IMG)      // 25600
#define NHEADS  8
#define HD      32
#define PP      4
#define WPATCH  40
#define NPATCH  (WPATCH * WPATCH)  // 1600
#define SCALE   0.17677669529663687f  // 1/sqrt(32)
#define LDSS    258                   // conv LDS row stride (floats)
#define OSTR    20                    // attn out-tile LDS stride (floats)

__device__ __forceinline__ v8f wmma_f32(v2f a, v2f b, v8f c) {
  // D(16x16,f32) = A(16x4,f32) * B(4x16,f32) + C
  return __builtin_amdgcn_wmma_f32_16x16x4_f32(
      /*neg_a=*/false, a, /*neg_b=*/false, b,
      /*c_mod=*/(short)0, c, /*reuse_a=*/false, /*reuse_b=*/false);
}

// ---------------------------------------------------------------------------
// Kernel 1: fused qkv (per-channel scale+bias) + windowed attention.
// One wave per (batch, head, window). Writes attention output in [b,c,h,w]
// layout into d_out (later consumed in-place by the 1x1 conv kernel).
// ---------------------------------------------------------------------------
__global__ __launch_bounds__(256)
void PatchSA_attn_kernel(const float* __restrict__ x,
                         const float* __restrict__ qkv_w,
                         const float* __restrict__ qkv_b,
                         const float* __restrict__ pos_encode,
                         const int*   __restrict__ pos_index,
                         float* __restrict__ y) {
  __shared__ float pbuf[8 * 16 * 17];        // per-wave 16x16 P tile (padded)
  __shared__ float obuf[8 * HD * OSTR];      // per-wave 32ch x 16px out tile

  const int lane = threadIdx.x & 31;
  const int wv   = threadIdx.x >> 5;
  const int half = lane >> 4;   // which 16-lane half
  const int l16  = lane & 15;

  const int u     = blockIdx.x * 8 + wv;          // (b, head, window) unit
  const int b_i   = u / (NHEADS * NPATCH);
  const int rem   = u - b_i * (NHEADS * NPATCH);
  const int head  = rem / NPATCH;
  const int patch = rem - head * NPATCH;
  const int py = patch / WPATCH, px = patch - py * WPATCH;
  const int y0 = py * PP, x0 = px * PP;

  const int    cb = head * HD;
  const float* xb = x + (size_t)b_i * DIM * HW;

  // This lane's token for the Q/K fragments (A 16x4 and B 4x16 share the
  // same (lane -> token, K-parity) mapping per ISA 7.12.2).
  const int pixQ = (y0 + (l16 >> 2)) * IMG + x0 + (l16 & 3);

  v2f aq[8], bk[8];
#pragma unroll
  for (int kk = 0; kk < 8; ++kk) {
#pragma unroll
    for (int t = 0; t < 2; ++t) {
      const int   d  = 4 * kk + 2 * half + t;     // head-dim index
      const int   c  = cb + d;
      const float xv = xb[(size_t)c * HW + pixQ]; // one x read feeds q and k
      aq[kk][t] = fmaf(xv, qkv_w[c], qkv_b[c]) * SCALE;       // q * scale
      bk[kk][t] = fmaf(xv, qkv_w[DIM + c], qkv_b[DIM + c]);   // k
    }
  }

  // S = (scale*Q) * K^T : chain 8 K=4 WMMAs
  v8f s = {0.f, 0.f, 0.f, 0.f, 0.f, 0.f, 0.f, 0.f};
#pragma unroll
  for (int kk = 0; kk < 8; ++kk) s = wmma_f32(aq[kk], bk[kk], s);

  // Bias + softmax. C-layout: element (row = r + 8*half, col = l16) in s[r].
  float p[8];
#pragma unroll
  for (int r = 0; r < 8; ++r) {
    const int   i    = r + 8 * half;
    const float bias = pos_encode[pos_index[i * 16 + l16] * NHEADS + head];
    float v = s[r] + bias;
    float m = v;
#pragma unroll
    for (int off = 8; off >= 1; off >>= 1) m = fmaxf(m, __shfl_xor(m, off));
    const float e = __expf(v - m);
    float sum = e;
#pragma unroll
    for (int off = 8; off >= 1; off >>= 1) sum += __shfl_xor(sum, off);
    p[r] = e * __builtin_amdgcn_rcpf(sum);
  }

  // Re-shape P from C-layout to A-fragment layout through LDS.
  float* pb = pbuf + wv * (16 * 17);
#pragma unroll
  for (int r = 0; r < 8; ++r) pb[(r + 8 * half) * 17 + l16] = p[r];
  asm volatile("s_wait_dscnt 0" ::: "memory");  // wave-local cross-lane LDS dep

  v2f pa[4];
#pragma unroll
  for (int kk = 0; kk < 4; ++kk) {
#pragma unroll
    for (int t = 0; t < 2; ++t) {
      const int j = 4 * kk + 2 * half + t;
      pa[kk][t] = pb[l16 * 17 + j];
    }
  }

  // O(16x32) = P(16x16) * V(16x32), two 16-wide N halves.
  float* ob = obuf + wv * (HD * OSTR);
#pragma unroll
  for (int hN = 0; hN < 2; ++hN) {
    const int   cl  = hN * 16 + l16;       // lane's channel within head
    const int   c   = cb + cl;
    const float wv_ = qkv_w[2 * DIM + c];
    const float bv_ = qkv_b[2 * DIM + c];
    v2f bv[4];
#pragma unroll
    for (int kk = 0; kk < 4; ++kk) {
#pragma unroll
      for (int t = 0; t < 2; ++t) {
        const int j   = 4 * kk + 2 * half + t;           // token (K dim)
        const int pix = (y0 + (j >> 2)) * IMG + x0 + (j & 3);
        bv[kk][t] = fmaf(xb[(size_t)c * HW + pix], wv_, bv_);
      }
    }
    v8f o = {0.f, 0.f, 0.f, 0.f, 0.f, 0.f, 0.f, 0.f};
#pragma unroll
    for (int kk = 0; kk < 4; ++kk) o = wmma_f32(pa[kk], bv[kk], o);

    // Park this half's D tile in LDS: 8 consecutive floats per lane (b128s).
#pragma unroll
    for (int r = 0; r < 8; ++r) ob[cl * OSTR + 8 * half + r] = o[r];
  }
  asm volatile("s_wait_dscnt 0" ::: "memory");

  // 16B-contiguous stores: each (channel, window-row) is 4 aligned floats.
#pragma unroll
  for (int it = 0; it < 4; ++it) {
    const int seg = it * 32 + lane;   // 0..127 = 32 channels x 4 rows
    const int cl  = seg >> 2;
    const int dy  = seg & 3;
    const v4f vv  = *(const v4f*)(ob + cl * OSTR + dy * 4);
    *(v4f*)(y + ((size_t)b_i * DIM + cb + cl) * HW +
            (size_t)(y0 + dy) * IMG + x0) = vv;
  }
}

// ---------------------------------------------------------------------------
// Kernel 2: 1x1 conv (out = W[256x256] x y) done IN PLACE on d_out.
// Each pixel's outputs depend only on that pixel's 256 input channels, so a
// block stages its 32 pixels x 256 channels into LDS (via ASYNC-to-LDS with
// a transposed per-lane LDS scatter), then overwrites them.
// GEMM tiling: 8 waves x (2 o-tiles x 2 pixel-tiles), K=256 as 64 x4-chunks.
// ---------------------------------------------------------------------------
__global__ __launch_bounds__(256)
void PatchSA_conv_kernel(const float* __restrict__ Wm, float* __restrict__ y) {
  __shared__ float ybuf[32 * LDSS];  // [pixel][channel]; sole LDS object -> base 0

  const int t    = threadIdx.x;
  const int lane = t & 31, wv = t >> 5;
  const int half = lane >> 4, l16 = lane & 15;

  const int g0   = blockIdx.x * 32;       // flat pixel over (b, h*w)
  const int b_i  = g0 / HW;               // 32 | HW -> no straddle
  const int pix0 = g0 - b_i * HW;
  float* yb = y + (size_t)b_i * DIM * HW;

  const int o0 = (2 * wv) * 16 + l16;     // A-fragment rows (M = l16)
  const int o1 = o0 + 16;
  __builtin_prefetch(Wm + (size_t)o0 * DIM, 0, 0);  // warm L2 weight rows
  __builtin_prefetch(Wm + (size_t)o1 * DIM, 0, 0);

  // Stage: coalesced 128B global reads async-DMA'd straight into LDS with a
  // per-lane transposed destination (no VGPR round trip, ASYNCcnt-tracked).
#pragma unroll 4
  for (int it = 0; it < 32; ++it) {
    const int      c        = it * 8 + wv;
    const float*   g        = yb + (size_t)c * HW + pix0 + lane;
    const unsigned lds_byte = (unsigned)((lane * LDSS + c) * 4);
    asm volatile("global_load_async_to_lds_b32 %0, %1, off"
                 :: "v"(lds_byte), "v"(g) : "memory");
  }
  asm volatile("s_wait_asynccnt 0" ::: "memory");
  __syncthreads();

  v8f acc0 = {0.f,0.f,0.f,0.f,0.f,0.f,0.f,0.f};
  v8f acc1 = acc0, acc2 = acc0, acc3 = acc0;

#pragma unroll 4
  for (int kk = 0; kk < 64; ++kk) {
    const int c0 = 4 * kk + 2 * half;     // K-pair this lane owns (even -> 8B aligned)
    const v2f A0 = *(const v2f*)(Wm + (size_t)o0 * DIM + c0);   // L2-resident
    const v2f A1 = *(const v2f*)(Wm + (size_t)o1 * DIM + c0);
    const v2f B0 = *(const v2f*)(ybuf + l16 * LDSS + c0);
    const v2f B1 = *(const v2f*)(ybuf + (16 + l16) * LDSS + c0);
    acc0 = wmma_f32(A0, B0, acc0);
    acc1 = wmma_f32(A0, B1, acc1);
    acc2 = wmma_f32(A1, B0, acc2);
    acc3 = wmma_f32(A1, B1, acc3);
  }

  // Stores: lanes 0-15 / 16-31 each write 64B-contiguous pixel runs.
#pragma unroll
  for (int r = 0; r < 8; ++r) {
    const int    mrow = r + 8 * half;
    const size_t rowA = ((size_t)b_i * DIM + 2 * wv * 16 + mrow) * HW + pix0;
    const size_t rowB = ((size_t)b_i * DIM + (2 * wv + 1) * 16 + mrow) * HW + pix0;
    y[rowA + l16]      = acc0[r];
    y[rowA + 16 + l16] = acc1[r];
    y[rowB + l16]      = acc2[r];
    y[rowB + 16 + l16] = acc3[r];
  }
}

extern "C" void kernel_launch(void* const* d_in, const int* in_sizes, int n_in,
                              void* d_out, int out_size, void* d_ws, size_t ws_size,
                              hipStream_t stream) {
  (void)in_sizes; (void)n_in; (void)out_size; (void)d_ws; (void)ws_size;
  const float* x          = (const float*)d_in[0];
  const float* qkv_w      = (const float*)d_in[1];
  const float* qkv_b      = (const float*)d_in[2];
  const float* pos_encode = (const float*)d_in[3];
  const float* out_w      = (const float*)d_in[4];
  const int*   pos_index  = (const int*)d_in[5];
  float* out = (float*)d_out;

  // 8*8*1600 = 102400 attention units, 8 waves (256 threads) per block.
  PatchSA_attn_kernel<<<(NB * NHEADS * NPATCH) / 8, 256, 0, stream>>>(
      x, qkv_w, qkv_b, pos_encode, pos_index, out);

  // 204800 pixels / 32 per block = 6400 blocks; in-place 1x1 conv on d_out.
  PatchSA_conv_kernel<<<(NB * HW) / 32, 256, 0, stream>>>(out_w, out);
}